// BlockTrainerBlend_27977416966710
// MI455X (gfx1250) — compile-verified
//
#include <hip/hip_runtime.h>
#include <hip/hip_bf16.h>

typedef float v2f __attribute__((ext_vector_type(2)));
typedef float v8f __attribute__((ext_vector_type(8)));

#define B_SZ   2048
#define MM_D   1600
#define C_CH   20
#define S_CH   80
#define R_RK   5
#define O_DIM  27
#define NRS    400   // R*S
#define ROWS   64    // batch rows per stage-1 block
#define MT     4     // ROWS/16 M-tiles
#define ST     5     // S_CH/16 s-tiles
#define NJOBS  20    // MT*ST

// Stage 1: per (chunk, 64-row group): fused GEMM -> rank-sum -> signed-sqrt ->
// L2-normalize -> scaled dots z_c . Wo_c written to the chunks_out region
// of d_out (raw logit contributions; softmaxed in stage 2).
__global__ __launch_bounds__(256)
void stage1(const float* __restrict__ x0, const float* __restrict__ x1,
            const float* __restrict__ x2, const float* __restrict__ W,
            const float* __restrict__ bias, const float* __restrict__ Wout,
            float* __restrict__ dots /* [B,C,O] staging (== chunks_out) */)
{
    __shared__ float zbuf[ROWS][S_CH];   // 20.5 KB
    __shared__ float scaleb[ROWS];

    const int tid  = threadIdx.x;
    const int lane = tid & 31;
    const int wave = tid >> 5;
    const int half = lane >> 4;          // A/B fragment K-half (K+0/1 vs K+2/3)
    const int l15  = lane & 15;
    const int c        = blockIdx.x;     // chunk
    const int rowBase0 = blockIdx.y * ROWS;

    const float* xs[3] = { x0, x1, x2 };

    // -------- GEMM + in-register rank/modality fusion --------
    for (int job = wave; job < NJOBS; job += 8) {
        const int mt = job / ST;               // M-tile within the 64 rows
        const int st = job % ST;               // s-tile (z columns)
        const int rowBase = rowBase0 + mt * 16;
        const int sb = st * 16 + l15;          // this lane's z column in [0,80)

        v8f pr0, pr1, pr2, pr3, pr4;           // running product over modalities
        #pragma unroll
        for (int m = 0; m < 3; ++m) {
            // 5 independent accumulators (rank r), bias folded into C init
            v8f acc[R_RK];
            #pragma unroll
            for (int r = 0; r < R_RK; ++r) {
                const float bv = bias[(m * C_CH + c) * NRS + r * S_CH + sb];
                acc[r] = (v8f){bv, bv, bv, bv, bv, bv, bv, bv};
            }
            // A frag: row = l15 of this M-tile, k = k0 + 2*half (+1)
            const float* Ap = xs[m] + (size_t)(rowBase + l15) * MM_D
                                    + c * S_CH + 2 * half;
            // B frag: B[k][n] = W[n][k]; rank r adds a constant row offset
            const float* Bp = W + ((size_t)(m * C_CH + c) * NRS + sb) * S_CH
                                + 2 * half;
            #pragma unroll 2
            for (int k0 = 0; k0 < S_CH; k0 += 4) {
                const v2f a = *(const v2f*)(Ap + k0);     // shared by 5 WMMAs
                #pragma unroll
                for (int r = 0; r < R_RK; ++r) {
                    const v2f bfr = *(const v2f*)(Bp + k0 + r * (S_CH * S_CH));
                    acc[r] = __builtin_amdgcn_wmma_f32_16x16x4_f32(
                                 false, a, false, bfr, (short)0, acc[r],
                                 false, false);
                }
            }
            if (m == 0) { pr0 = acc[0]; pr1 = acc[1]; pr2 = acc[2];
                          pr3 = acc[3]; pr4 = acc[4]; }
            else        { pr0 *= acc[0]; pr1 *= acc[1]; pr2 *= acc[2];
                          pr3 *= acc[3]; pr4 *= acc[4]; }
        }
        const v8f z = pr0 + pr1 + pr2 + pr3 + pr4;   // sum over rank
        #pragma unroll
        for (int j = 0; j < 8; ++j) {                // C/D layout: VGPR j -> M=j+8*half
            const float v = z[j];
            zbuf[mt * 16 + j + half * 8][sb] =
                sqrtf(fmaxf(v, 0.f)) - sqrtf(fmaxf(-v, 0.f));
        }
    }
    __syncthreads();

    // -------- per-row L2 scale --------
    if (tid < ROWS) {
        float ss = 0.f;
        for (int s = 0; s < S_CH; ++s) { const float v = zbuf[tid][s]; ss += v * v; }
        scaleb[tid] = 1.f / fmaxf(sqrtf(ss), 1e-12f);
    }
    __syncthreads();

    // -------- scaled dots: dot[b][c][o] = scale_b * (z_b . Wo[o,c,:]) --------
    for (int i = tid; i < ROWS * O_DIM; i += 256) {
        const int br = i / O_DIM, o = i % O_DIM;
        const float* wo = Wout + (size_t)o * MM_D + c * S_CH;
        float acc = 0.f;
        for (int s = 0; s < S_CH; ++s) acc += zbuf[br][s] * wo[s];
        dots[((size_t)(rowBase0 + br) * C_CH + c) * O_DIM + o] = acc * scaleb[br];
    }
}

// Stage 2: one thread per batch row. Converts raw dots -> per-chunk softmax
// (in place) and accumulates final logits = sum_c dot + b_out -> final softmax.
__global__ __launch_bounds__(256)
void stage2(float* __restrict__ dots,          // [B,C,O] in/out (chunks_out)
            const float* __restrict__ bout,    // [O]
            float* __restrict__ out_final)     // [B,O]
{
    const int b = blockIdx.x * blockDim.x + threadIdx.x;
    if (b >= B_SZ) return;

    float fin[O_DIM];
    #pragma unroll
    for (int o = 0; o < O_DIM; ++o) fin[o] = bout[o];

    float* row = dots + (size_t)b * C_CH * O_DIM;
    for (int c = 0; c < C_CH; ++c) {
        float* d = row + c * O_DIM;
        float l[O_DIM];
        float mx = -1e30f;
        #pragma unroll
        for (int o = 0; o < O_DIM; ++o) {
            const float dv = d[o];
            fin[o] += dv;
            l[o] = dv + bout[o];
            mx = fmaxf(mx, l[o]);
        }
        float se = 0.f;
        #pragma unroll
        for (int o = 0; o < O_DIM; ++o) { l[o] = expf(l[o] - mx); se += l[o]; }
        const float inv = 1.f / se;
        #pragma unroll
        for (int o = 0; o < O_DIM; ++o) d[o] = l[o] * inv;
    }

    float mx = -1e30f;
    #pragma unroll
    for (int o = 0; o < O_DIM; ++o) mx = fmaxf(mx, fin[o]);
    float se = 0.f;
    #pragma unroll
    for (int o = 0; o < O_DIM; ++o) { fin[o] = expf(fin[o] - mx); se += fin[o]; }
    const float inv = 1.f / se;
    float* dst = out_final + (size_t)b * O_DIM;
    #pragma unroll
    for (int o = 0; o < O_DIM; ++o) dst[o] = fin[o] * inv;
}

extern "C" void kernel_launch(void* const* d_in, const int* in_sizes, int n_in,
                              void* d_out, int out_size, void* d_ws, size_t ws_size,
                              hipStream_t stream) {
    const float* x0   = (const float*)d_in[0];
    const float* x1   = (const float*)d_in[1];
    const float* x2   = (const float*)d_in[2];
    const float* W    = (const float*)d_in[3];
    const float* bias = (const float*)d_in[4];
    const float* Wout = (const float*)d_in[5];
    const float* bout = (const float*)d_in[6];

    float* out_final  = (float*)d_out;                           // [B,27]
    float* out_chunks = (float*)d_out + (size_t)B_SZ * O_DIM;    // [B,C,27]

    dim3 g1(C_CH, B_SZ / ROWS);   // (20 chunks, 32 row-groups) = 640 blocks
    stage1<<<g1, 256, 0, stream>>>(x0, x1, x2, W, bias, Wout, out_chunks);
    stage2<<<B_SZ / 256, 256, 0, stream>>>(out_chunks, bout, out_final);
}